// VGCNBlock_10247791968469
// MI455X (gfx1250) — compile-verified
//
#include <hip/hip_runtime.h>
#include <hip/hip_bf16.h>

#define N_NODES 50000
#define N_EDGES 640000
#define DIM 128
#define ALPHA 0.5f

typedef __attribute__((ext_vector_type(2))) float v2f;
typedef __attribute__((ext_vector_type(8))) float v8f;

// ---------------- degree ----------------
__global__ void k_degree(const int* __restrict__ dst, int* __restrict__ degi, int E) {
    int e = blockIdx.x * blockDim.x + threadIdx.x;
    if (e < E) atomicAdd(&degi[dst[e]], 1);
}

// ---------------- norm = (clip(deg,1)+1)^-0.5 ----------------
__global__ void k_norm(const int* __restrict__ degi, float* __restrict__ norm, int N) {
    int i = blockIdx.x * blockDim.x + threadIdx.x;
    if (i >= N) return;
    float d = (float)(degi[i] < 1 ? 1 : degi[i]);
    norm[i] = rsqrtf(d + 1.0f);
}

// ---------------- per-edge L1 distance + segment max (one wave32 per edge) ----------------
__global__ void __launch_bounds__(256) k_l1(const float* __restrict__ F,
                                            const int* __restrict__ src,
                                            const int* __restrict__ dst,
                                            float* __restrict__ ebuf,
                                            int* __restrict__ emaxi, int E) {
    int e = blockIdx.x * 8 + (threadIdx.x >> 5);
    if (e >= E) return;
    int lane = threadIdx.x & 31;
    int s = src[e], d = dst[e];
    const float4 fs = *reinterpret_cast<const float4*>(F + (size_t)s * DIM + lane * 4);
    const float4 fd = *reinterpret_cast<const float4*>(F + (size_t)d * DIM + lane * 4);
    float t = fabsf(fs.x - fd.x) + fabsf(fs.y - fd.y) + fabsf(fs.z - fd.z) + fabsf(fs.w - fd.w);
#pragma unroll
    for (int off = 16; off > 0; off >>= 1) t += __shfl_xor(t, off, 32);
    if (lane == 0) {
        ebuf[e] = t;
        // l1 >= 0 always -> int-bit max is monotone; emax initialized to 0.0f bits
        atomicMax(&emaxi[d], __float_as_int(t));
    }
}

// ---------------- ex = exp(l1 - emax[dst]); esum += ex (in-place on ebuf) ----------------
__global__ void k_ex(const float* __restrict__ emaxf, const int* __restrict__ dst,
                     float* __restrict__ ebuf, float* __restrict__ esum, int E) {
    int e = blockIdx.x * blockDim.x + threadIdx.x;
    if (e >= E) return;
    int d = dst[e];
    float ex = expf(ebuf[e] - emaxf[d]);
    ebuf[e] = ex;
    atomicAdd(&esum[d], ex);
}

// ---------------- dense GEMM  h = F * W^T + b  via V_WMMA_F32_16X16X4_F32 ----------------
// Block = 256 threads = 8 waves. Block owns a 16-row node tile (A), staged into LDS with
// GLOBAL_LOAD_ASYNC_TO_LDS_B128 (ASYNCcnt-tracked direct global->LDS DMA, no VGPR bounce);
// wave w owns output columns [w*16, w*16+16). K-loop: 32 x WMMA 16x16x4 (K=128).
// A-matrix VGPR layout (32-bit, 16x4): lane L -> row M = L%16; VGPR0 = K {0|2}, VGPR1 = K {1|3}
// (upper half-wave holds K+2). B 4x16 symmetric: lane L -> col N = L%16, K offset (L/16)*2.
// C/D 16x16: VGPR v, lane L -> M = v + (L/16)*8, N = L%16.
__global__ void __launch_bounds__(256) k_gemm(const float* __restrict__ F,
                                              const float* __restrict__ W,
                                              const float* __restrict__ bias,
                                              const float* __restrict__ norm,
                                              float* __restrict__ h_pre,
                                              float* __restrict__ ri) {
    __shared__ float As[16 * DIM];  // 8 KB
    const int tid = threadIdx.x;
    const int m0 = blockIdx.x * 16;

    // Async A-tile stage: 2048 floats = 512 x b128; 256 lanes x 2 transfers.
    // VDST = 32-bit LDS byte address (low 32 bits of the generic shared pointer);
    // VADDR = 64-bit global address; saddr = off. Tracked by ASYNCcnt.
    {
        const float4* gsrc = reinterpret_cast<const float4*>(F + (size_t)m0 * DIM);
        unsigned lds_base = (unsigned)(uintptr_t)As;  // LDS aperture: addr[31:0] = LDS offset
        unsigned la0 = lds_base + tid * 16;
        unsigned la1 = lds_base + (tid + 256) * 16;
        unsigned long long ga0 = (unsigned long long)(uintptr_t)(gsrc + tid);
        unsigned long long ga1 = (unsigned long long)(uintptr_t)(gsrc + tid + 256);
        asm volatile("global_load_async_to_lds_b128 %0, %1, off"
                     :: "v"(la0), "v"(ga0) : "memory");
        asm volatile("global_load_async_to_lds_b128 %0, %1, off"
                     :: "v"(la1), "v"(ga1) : "memory");
    }

    const int wave = tid >> 5;
    const int lane = tid & 31;
    const int n0 = wave * 16;
    const int row = lane & 15;         // A: M, B: N
    const int khalf = (lane >> 4) * 2; // 0 or 2

    // Preload B operands: W[(n0+row)][k0+khalf .. +1] for k0 = 0,4,...,124 (W row-major, hot
    // in L2) — overlaps with the in-flight async LDS fill above.
    v2f bv[32];
    const float* wrow = W + (size_t)(n0 + row) * DIM + khalf;
#pragma unroll
    for (int kk = 0; kk < 32; ++kk) {
        bv[kk].x = wrow[kk * 4 + 0];
        bv[kk].y = wrow[kk * 4 + 1];
    }

    // Drain the async tile fill, then workgroup barrier before any wave touches As.
    asm volatile("s_wait_asynccnt 0" ::: "memory");
    __syncthreads();

    v8f c = {};
    const float* arow = As + row * DIM + khalf;
#pragma unroll
    for (int kk = 0; kk < 32; ++kk) {
        v2f a;
        a.x = arow[kk * 4 + 0];
        a.y = arow[kk * 4 + 1];
        // 8 args: (neg_a, A, neg_b, B, c_mod, C, reuse_a, reuse_b)
        c = __builtin_amdgcn_wmma_f32_16x16x4_f32(false, a, false, bv[kk], (short)0, c,
                                                  false, false);
    }

#pragma unroll
    for (int v = 0; v < 8; ++v) {
        int node = m0 + v + (lane >> 4) * 8;
        int col = n0 + (lane & 15);
        float val = c[v] + bias[col];
        float nn = norm[node];
        size_t idx = (size_t)node * DIM + col;
        h_pre[idx] = val;           // h (pre-propagation) and h_pre
        ri[idx] = val * nn * nn;    // residual term ri = h * norm^2
    }
}

// ---------------- propagation: acc[dst] += h_pre[src] * norm[src] * (ex/esum[dst]) ----------------
__global__ void __launch_bounds__(256) k_prop(const float* __restrict__ h,
                                              const float* __restrict__ norm,
                                              const float* __restrict__ ebuf,
                                              const float* __restrict__ esum,
                                              const int* __restrict__ src,
                                              const int* __restrict__ dst,
                                              float* __restrict__ acc, int E) {
    int e = blockIdx.x * 8 + (threadIdx.x >> 5);
    if (e >= E) return;
    int lane = threadIdx.x & 31;
    int s = src[e], d = dst[e];
    float a = (ebuf[e] / esum[d]) * norm[s];
    const float4 hv = *reinterpret_cast<const float4*>(h + (size_t)s * DIM + lane * 4);
    float* ap = acc + (size_t)d * DIM + lane * 4;
    atomicAdd(ap + 0, hv.x * a);
    atomicAdd(ap + 1, hv.y * a);
    atomicAdd(ap + 2, hv.z * a);
    atomicAdd(ap + 3, hv.w * a);
}

// ---------------- update: h = a*acc*norm + a*ri + (1-a)*h_pre ; final -> ReLU to out ----------------
__global__ void k_update(float* __restrict__ h_pre, const float* __restrict__ ri,
                         float* __restrict__ acc, const float* __restrict__ norm,
                         float* __restrict__ out, int last, int total) {
    int i = blockIdx.x * blockDim.x + threadIdx.x;
    if (i >= total) return;
    int node = i >> 7;  // DIM == 128
    float val = ALPHA * acc[i] * norm[node] + ALPHA * ri[i] + (1.0f - ALPHA) * h_pre[i];
    if (last) {
        out[i] = fmaxf(val, 0.0f);
    } else {
        h_pre[i] = val;
        acc[i] = 0.0f;  // reset accumulator for the next round
    }
}

extern "C" void kernel_launch(void* const* d_in, const int* in_sizes, int n_in,
                              void* d_out, int out_size, void* d_ws, size_t ws_size,
                              hipStream_t stream) {
    const float* F    = (const float*)d_in[0];
    const float* W    = (const float*)d_in[1];
    const float* bias = (const float*)d_in[2];
    const int*   src  = (const int*)d_in[3];
    const int*   dst  = (const int*)d_in[4];
    float* out = (float*)d_out;

    const int N = N_NODES, E = N_EDGES;
    char* ws = (char*)d_ws;
    int*   degi  = (int*)ws;                 ws += (size_t)N * sizeof(int);
    float* norm  = (float*)ws;               ws += (size_t)N * sizeof(float);
    int*   emaxi = (int*)ws;                 ws += (size_t)N * sizeof(int);
    float* esum  = (float*)ws;               ws += (size_t)N * sizeof(float);
    float* ebuf  = (float*)ws;               ws += (size_t)E * sizeof(float);  // l1 -> ex
    float* ri    = (float*)ws;               ws += (size_t)N * DIM * sizeof(float);
    float* h_pre = (float*)ws;               ws += (size_t)N * DIM * sizeof(float);
    float* acc   = out;  // d_out doubles as the scatter accumulator

    hipMemsetAsync(degi, 0, (size_t)N * sizeof(int), stream);
    hipMemsetAsync(emaxi, 0, (size_t)N * sizeof(int), stream);   // 0.0f bits; l1 >= 0
    hipMemsetAsync(esum, 0, (size_t)N * sizeof(float), stream);
    hipMemsetAsync(acc, 0, (size_t)N * DIM * sizeof(float), stream);

    k_degree<<<(E + 255) / 256, 256, 0, stream>>>(dst, degi, E);
    k_norm<<<(N + 255) / 256, 256, 0, stream>>>(degi, norm, N);
    k_l1<<<E / 8, 256, 0, stream>>>(F, src, dst, ebuf, emaxi, E);            // 640000 % 8 == 0
    k_ex<<<(E + 255) / 256, 256, 0, stream>>>((const float*)emaxi, dst, ebuf, esum, E);
    k_gemm<<<N / 16, 256, 0, stream>>>(F, W, bias, norm, h_pre, ri);         // 50000 % 16 == 0

    for (int k = 0; k < 2; ++k) {
        k_prop<<<E / 8, 256, 0, stream>>>(h_pre, norm, ebuf, esum, src, dst, acc, E);
        k_update<<<(N * DIM + 255) / 256, 256, 0, stream>>>(h_pre, ri, acc, norm, out,
                                                            (k == 1) ? 1 : 0, N * DIM);
    }
}